// RGCN_36773509988955
// MI455X (gfx1250) — compile-verified
//
#include <hip/hip_runtime.h>

typedef __attribute__((ext_vector_type(2))) float v2f;
typedef __attribute__((ext_vector_type(8))) float v8f;

#define D   128   // feature dim (fixed by problem)
#define TM  128   // rows per block tile
#define KP  32    // K panel staged in LDS
#define XS  36    // padded sX row stride (floats): conflict-free A-frag b64 loads
#define WS  288   // padded packed-W pair-row stride (floats): disjoint banks per lane-half

// ---------------------------------------------------------------------------
// Dense transform: O = X @ W for three weight matrices (W_rel0, W_rel1, root),
// selected by blockIdx.y. fp32 WMMA (16x16x4) keeps full reference precision.
// W staged in LDS pre-interleaved as (k,k+1) pairs so every B fragment is one
// aligned ds_load_b64 feeding v_wmma directly.
// ---------------------------------------------------------------------------
__global__ __launch_bounds__(256) void rgcn_gemm(
    const float* __restrict__ X,
    const float* __restrict__ Wa,
    const float* __restrict__ Wb,
    const float* __restrict__ Wr,
    float* __restrict__ Oa,
    float* __restrict__ Ob,
    float* __restrict__ Or,
    int N)
{
    __shared__ float sX[TM * XS];          // 18 KB: X panel  [row][k], padded
    __shared__ float sWp[(KP / 2) * WS];   // 18 KB: W panel, pair-interleaved

    const float* W = (blockIdx.y == 0) ? Wa : ((blockIdx.y == 1) ? Wb : Wr);
    float*       O = (blockIdx.y == 0) ? Oa : ((blockIdx.y == 1) ? Ob : Or);

    const int tid  = threadIdx.x;
    const int wave = tid >> 5;
    const int lane = tid & 31;
    const int half = lane >> 4;   // 0: lanes 0-15, 1: lanes 16-31
    const int l16  = lane & 15;
    const long long rowBase = (long long)blockIdx.x * TM;

    v8f acc[8] = {};  // 8 col-tiles of 16x16 f32 accumulators

    for (int kp = 0; kp < D; kp += KP) {
        // --- cooperative load: X panel (clamped rows for the tail block) ---
        {
            const int r  = tid >> 3;            // 0..31
            const int cg = (tid & 7) * 4;       // 0..28
            #pragma unroll
            for (int it = 0; it < 4; ++it) {
                const int row = r + 32 * it;
                long long gr = rowBase + row;
                if (gr > (long long)N - 1) gr = (long long)N - 1;
                *(float4*)(sX + row * XS + cg) =
                    *(const float4*)(X + gr * D + kp + cg);
            }
        }
        // --- cooperative load: W panel, interleaving rows k and k+1 ---
        {
            const int p = tid >> 4;             // pair index 0..15
            const int g = (tid & 15) * 8;       // col group 0..120
            const long long ra = (long long)(kp + 2 * p) * D;
            const float4 va0 = *(const float4*)(W + ra + g);
            const float4 va1 = *(const float4*)(W + ra + g + 4);
            const float4 vb0 = *(const float4*)(W + ra + D + g);
            const float4 vb1 = *(const float4*)(W + ra + D + g + 4);
            float4* dst = (float4*)(sWp + p * WS + g * 2);
            dst[0] = make_float4(va0.x, vb0.x, va0.y, vb0.y);
            dst[1] = make_float4(va0.z, vb0.z, va0.w, vb0.w);
            dst[2] = make_float4(va1.x, vb1.x, va1.y, vb1.y);
            dst[3] = make_float4(va1.z, vb1.z, va1.w, vb1.w);
        }
        __syncthreads();

        // --- WMMA: 8 K-steps x 8 col-tiles per panel ---
        const int m = wave * 16 + l16;          // A-matrix row (M = lane % 16)
        #pragma unroll
        for (int ks = 0; ks < KP; ks += 4) {
            // ISA layout: lanes 0-15 carry K={ks,ks+1}, lanes 16-31 K={ks+2,ks+3}
            const v2f a = *(const v2f*)(sX + m * XS + ks + 2 * half);
            const float* brow = sWp + ((ks >> 1) + half) * WS;
            #pragma unroll
            for (int c = 0; c < 8; ++c) {
                const v2f b = *(const v2f*)(brow + (c * 16 + l16) * 2);
                acc[c] = __builtin_amdgcn_wmma_f32_16x16x4_f32(
                    false, a, false, b, (short)0, acc[c], false, false);
            }
        }
        __syncthreads();
    }

    // --- store: C/D layout VGPR j -> M = j + 8*half, N = lane%16 ---
    #pragma unroll
    for (int c = 0; c < 8; ++c) {
        #pragma unroll
        for (int j = 0; j < 8; ++j) {
            const long long row = rowBase + wave * 16 + j + 8 * half;
            if (row < N) O[row * D + c * 16 + l16] = acc[c][j];
        }
    }
}

// ---------------------------------------------------------------------------
// Edge scatter: one wave per edge; 4 coalesced loads + 4 f32 HW atomics.
// sums layout: [(node*2 + rel)][128]  (102 MB -> L2-resident on MI455X)
// ---------------------------------------------------------------------------
__global__ __launch_bounds__(256) void rgcn_scatter(
    const long long* __restrict__ ei,   // [2, E]
    const float* __restrict__ xr,       // [N, 128] (already relation-selected)
    float* __restrict__ sums,
    int E, int rel)
{
    const long long t = (long long)blockIdx.x * blockDim.x + threadIdx.x;
    const long long e = t >> 5;
    const int lane = (int)(t & 31);
    if (e >= E) return;
    const long long src = ei[e];
    const long long dst = ei[(long long)E + e];
    const float* in  = xr + src * D;
    float*       out = sums + (dst * 2 + rel) * D;
    #pragma unroll
    for (int j = 0; j < 4; ++j) {
        const int f = lane + 32 * j;
        unsafeAtomicAdd(out + f, in[f]);
    }
}

__global__ __launch_bounds__(256) void rgcn_count(
    const long long* __restrict__ ei, float* __restrict__ cnt, int E, int rel)
{
    const long long e = (long long)blockIdx.x * blockDim.x + threadIdx.x;
    if (e >= E) return;
    const long long dst = ei[(long long)E + e];
    unsafeAtomicAdd(cnt + dst * 2 + rel, 1.0f);
}

__global__ __launch_bounds__(256) void rgcn_inv(float* __restrict__ cnt, int n)
{
    const long long i = (long long)blockIdx.x * blockDim.x + threadIdx.x;
    if (i < n) cnt[i] = 1.0f / fmaxf(cnt[i], 1.0f);
}

__global__ __launch_bounds__(256) void rgcn_zero(float4* __restrict__ p, long long n4)
{
    const long long i = (long long)blockIdx.x * blockDim.x + threadIdx.x;
    if (i < n4) p[i] = make_float4(0.f, 0.f, 0.f, 0.f);
}

// io[n,:] (pre-filled with x@root) += bias + sum_r sums[n,r,:]*inv[n,r]; opt. ReLU
__global__ __launch_bounds__(256) void rgcn_combine(
    const float* __restrict__ sums, const float* __restrict__ inv,
    const float* __restrict__ bias, float* __restrict__ io, int N, int relu)
{
    const long long t = (long long)blockIdx.x * blockDim.x + threadIdx.x;
    if (t >= (long long)N * 32) return;
    const long long n = t >> 5;
    const int q = (int)(t & 31) * 4;
    const float i0 = inv[n * 2];
    const float i1 = inv[n * 2 + 1];
    const float4 s0 = *(const float4*)(sums + (n * 2) * D + q);
    const float4 s1 = *(const float4*)(sums + (n * 2 + 1) * D + q);
    const float4 r  = *(const float4*)(io + n * D + q);
    const float4 b  = *(const float4*)(bias + q);
    float4 v;
    v.x = r.x + b.x + s0.x * i0 + s1.x * i1;
    v.y = r.y + b.y + s0.y * i0 + s1.y * i1;
    v.z = r.z + b.z + s0.z * i0 + s1.z * i1;
    v.w = r.w + b.w + s0.w * i0 + s1.w * i1;
    if (relu) {
        v.x = fmaxf(v.x, 0.f); v.y = fmaxf(v.y, 0.f);
        v.z = fmaxf(v.z, 0.f); v.w = fmaxf(v.w, 0.f);
    }
    *(float4*)(io + n * D + q) = v;
}

extern "C" void kernel_launch(void* const* d_in, const int* in_sizes, int n_in,
                              void* d_out, int out_size, void* d_ws, size_t ws_size,
                              hipStream_t stream) {
    const float*     x     = (const float*)d_in[0];
    const long long* ea    = (const long long*)d_in[1];
    const long long* eb    = (const long long*)d_in[2];
    const float*     W1    = (const float*)d_in[3];
    const float*     root1 = (const float*)d_in[4];
    const float*     b1    = (const float*)d_in[5];
    const float*     W2    = (const float*)d_in[6];
    const float*     root2 = (const float*)d_in[7];
    const float*     b2    = (const float*)d_in[8];
    float* out = (float*)d_out;

    const int N = in_sizes[0] / D;
    const int E = in_sizes[1] / 2;

    // workspace layout (floats)
    float* xr   = (float*)d_ws;                       // 2*N*D   (per-relation transforms)
    float* sums = xr   + (size_t)2 * N * D;           // 2*N*D   (segment sums)
    float* cnt  = sums + (size_t)2 * N * D;           // 2*N     (counts -> inverse counts)
    float* h    = cnt  + (size_t)2 * N;               // N*D     (hidden layer)

    const int eblk   = (E + 255) / 256;
    const int sblk   = (int)(((long long)E * 32 + 255) / 256);
    const int cblk   = (int)(((long long)N * 32 + 255) / 256);
    const long long sums4 = (long long)2 * N * D / 4;
    const int zblk   = (int)((sums4 + 255) / 256);
    dim3 ggrid((N + TM - 1) / TM, 3);

    // --- topology-only counts (shared by both layers) ---
    {
        const long long c4 = ((long long)2 * N + 3) / 4;
        rgcn_zero<<<(int)((c4 + 255) / 256), 256, 0, stream>>>((float4*)cnt, c4);
    }
    rgcn_count<<<eblk, 256, 0, stream>>>(ea, cnt, E, 0);
    rgcn_count<<<eblk, 256, 0, stream>>>(eb, cnt, E, 1);
    rgcn_inv<<<(2 * N + 255) / 256, 256, 0, stream>>>(cnt, 2 * N);

    // --- layer 1 ---
    rgcn_zero<<<zblk, 256, 0, stream>>>((float4*)sums, sums4);
    rgcn_gemm<<<ggrid, 256, 0, stream>>>(x, W1, W1 + D * D, root1,
                                         xr, xr + (size_t)N * D, h, N);
    rgcn_scatter<<<sblk, 256, 0, stream>>>(ea, xr, sums, E, 0);
    rgcn_scatter<<<sblk, 256, 0, stream>>>(eb, xr + (size_t)N * D, sums, E, 1);
    rgcn_combine<<<cblk, 256, 0, stream>>>(sums, cnt, b1, h, N, 1);

    // --- layer 2 ---
    rgcn_zero<<<zblk, 256, 0, stream>>>((float4*)sums, sums4);
    rgcn_gemm<<<ggrid, 256, 0, stream>>>(h, W2, W2 + D * D, root2,
                                         xr, xr + (size_t)N * D, out, N);
    rgcn_scatter<<<sblk, 256, 0, stream>>>(ea, xr, sums, E, 0);
    rgcn_scatter<<<sblk, 256, 0, stream>>>(eb, xr + (size_t)N * D, sums, E, 1);
    rgcn_combine<<<cblk, 256, 0, stream>>>(sums, cnt, b2, out, N, 0);
}